// CLPLLoss_31688268709966
// MI455X (gfx1250) — compile-verified
//
#include <hip/hip_runtime.h>
#include <cstdint>
#include <cstddef>

// ---------------------------------------------------------------------------
// CLPL loss for MI455X (gfx1250): TDM-staged streaming reduction.
//   logits: [B, C] f32, cand_mask: [B, C] bool (1 byte each)
//   out: scalar f32 mean loss
// Memory-bound: 640MB @ 23.3TB/s ~= 27.5us floor. Logits are staged to LDS
// with tensor_load_to_lds (TENSORcnt, double-buffered, wave-0 driven); the
// mask streams via nontemporal b128 loads. wave32 shuffles for reductions.
// ---------------------------------------------------------------------------

typedef int          v4i_t __attribute__((ext_vector_type(4)));
typedef int          v8i_t __attribute__((ext_vector_type(8)));
typedef unsigned int v4u_t __attribute__((ext_vector_type(4)));

constexpr int C_CLS  = 32000;   // classes per row (fixed by reference)
constexpr int TILE   = 16000;   // floats per TDM tile (half row, 64 KB)
constexpr int BLOCK  = 512;     // 16 waves (wave32)
constexpr int NWAVES = BLOCK / 32;

__device__ inline float softplus_f(float x) {
    // stable: max(x,0) + log(1 + exp(-|x|)); v_exp_f32 / v_log_f32 fast path
    return fmaxf(x, 0.0f) + __logf(1.0f + __expf(-fabsf(x)));
}

// Issue one TDM load: TILE contiguous f32 from global -> LDS byte address.
// D# group0 (4 SGPRs) + group1 (8 SGPRs); 1-D tile, data_size=4B, no pad,
// no multicast (workgroup_mask=0: not in a cluster), no iterate.
__device__ inline void tdm_load_tile(const float* gsrc, uint32_t lds_byte_addr) {
    uint64_t ga = (uint64_t)(uintptr_t)gsrc;
    v4i_t g0;
    g0[0] = 1;                                   // count=1 (valid user D#)
    g0[1] = (int)lds_byte_addr;                  // lds_addr [63:32]
    g0[2] = (int)(uint32_t)ga;                   // global_addr lo
    g0[3] = (int)((uint32_t)((ga >> 32) & 0x01FFFFFFull) // global_addr[56:32]
                  | 0x80000000u);                // type=2 ("image") [127:126]
    v8i_t g1;
    g1[0] = 0x00020000;                          // data_size=2 (4 bytes)
    g1[1] = (int)(((uint32_t)TILE & 0xFFFFu) << 16);   // tensor_dim0 lo16 @[63:48]
    g1[2] = (int)((((uint32_t)TILE >> 16) & 0xFFFFu)   // tensor_dim0 hi16
                  | (1u << 16));                 // tensor_dim1 = 1
    g1[3] = (int)(((uint32_t)TILE & 0xFFFFu) << 16);   // tile_dim0 @[127:112]
    g1[4] = 0;                                   // tile_dim1=0 (1-D), tile_dim2=0
    g1[5] = TILE;                                // tensor_dim0_stride lo32
    g1[6] = 0;
    g1[7] = 0;
    asm volatile("tensor_load_to_lds %0, %1" :: "s"(g0), "s"(g1) : "memory");
}

__global__ __launch_bounds__(BLOCK) void clpl_rows_kernel(
    const float* __restrict__ logits,
    const unsigned char* __restrict__ mask,
    float* __restrict__ rowloss,
    int rows_per_block, int row_rem)
{
    __shared__ float buf[2][TILE];               // 128 KB double buffer
    __shared__ float red[NWAVES][4];

    const int tid = threadIdx.x;
    const int bid = blockIdx.x;
    const int G   = gridDim.x;
    const int nrows = rows_per_block + ((bid < row_rem) ? 1 : 0);
    const int nt = nrows * 2;                    // tiles (2 per row)
    if (nt == 0) return;

    auto issue = [&](int j) {
        const int row  = bid + (j >> 1) * G;
        const int half = j & 1;
        const float* src = logits + (size_t)row * C_CLS + (size_t)half * TILE;
        const uint32_t lds = (uint32_t)(uintptr_t)(void*)&buf[j & 1][0];
        tdm_load_tile(src, lds);
    };

    float sumAll = 0.f, sumPos = 0.f, sumNeg = 0.f, cntf = 0.f;

    if (tid < 32) issue(0);                      // wave 0 drives the TDM

    for (int j = 0; j < nt; ++j) {
        if (tid < 32) {
            if (j + 1 < nt) {
                issue(j + 1);                    // prefetch next tile
                __builtin_amdgcn_s_wait_tensorcnt(1);  // current tile done
            } else {
                __builtin_amdgcn_s_wait_tensorcnt(0);
            }
        }
        __syncthreads();                         // tile j visible to all waves

        const int row  = bid + (j >> 1) * G;
        const int half = j & 1;
        if (half == 0) { sumAll = sumPos = sumNeg = cntf = 0.f; }

        const float4* bp = (const float4*)&buf[j & 1][0];
        const v4u_t* mp = (const v4u_t*)(mask + (size_t)row * C_CLS
                                              + (size_t)half * TILE);
        // 16 elements (one b128 mask load + four b128 LDS reads) per iter
        for (int g = tid; g < TILE / 16; g += BLOCK) {
            const v4u_t mv = __builtin_nontemporal_load(mp + g);
#pragma unroll
            for (int w = 0; w < 4; ++w) {
                const float4 v = bp[g * 4 + w];  // ds_load_b128
                const uint32_t mw = mv[w];
                const float xs[4] = {v.x, v.y, v.z, v.w};
#pragma unroll
                for (int k = 0; k < 4; ++k) {
                    const float x = xs[k];
                    const bool m = ((mw >> (8 * k)) & 0xFFu) != 0u;
                    sumAll += x;
                    const float sp = softplus_f(x);
                    sumPos += m ? x   : 0.0f;
                    cntf   += m ? 1.f : 0.0f;
                    sumNeg += m ? 0.f : sp;
                }
            }
        }

        if (half == 1) {                         // row complete: reduce + epilogue
            float a = sumAll, p = sumPos, n = sumNeg, c = cntf;
            for (int o = 16; o > 0; o >>= 1) {   // wave32 tree
                a += __shfl_down(a, o, 32);
                p += __shfl_down(p, o, 32);
                n += __shfl_down(n, o, 32);
                c += __shfl_down(c, o, 32);
            }
            const int wid = tid >> 5;
            if ((tid & 31) == 0) {
                red[wid][0] = a; red[wid][1] = p; red[wid][2] = n; red[wid][3] = c;
            }
            __syncthreads();
            if (tid == 0) {
                float ta = 0.f, tp = 0.f, tn = 0.f, tc = 0.f;
                for (int w = 0; w < NWAVES; ++w) {
                    ta += red[w][0]; tp += red[w][1]; tn += red[w][2]; tc += red[w][3];
                }
                const bool empty = tc < 0.5f;    // cnt == 0
                const float pos  = empty ? (ta / (float)C_CLS) : (tp / tc);
                const float negc = (float)C_CLS - tc;
                const float negl = (!empty && negc > 0.5f) ? (tn / negc) : 0.0f;
                rowloss[row] = softplus_f(-pos) + negl;
            }
        }
        __syncthreads();                         // protect buf[j&1] before reuse
    }
}

__global__ void clpl_finalize_kernel(const float* __restrict__ rowloss,
                                     float* __restrict__ out, int B)
{
    __shared__ float red[32];
    float s = 0.f;
    for (int i = threadIdx.x; i < B; i += blockDim.x) s += rowloss[i];
    for (int o = 16; o > 0; o >>= 1) s += __shfl_down(s, o, 32);
    const int wid = threadIdx.x >> 5;
    if ((threadIdx.x & 31) == 0) red[wid] = s;
    __syncthreads();
    if (threadIdx.x == 0) {
        const int nw = (int)(blockDim.x + 31) / 32;
        float t = 0.f;
        for (int w = 0; w < nw; ++w) t += red[w];
        out[0] = t / (float)B;
    }
}

extern "C" void kernel_launch(void* const* d_in, const int* in_sizes, int n_in,
                              void* d_out, int out_size, void* d_ws, size_t ws_size,
                              hipStream_t stream) {
    (void)n_in; (void)out_size; (void)ws_size;
    const float*         logits = (const float*)d_in[0];
    const unsigned char* mask   = (const unsigned char*)d_in[1];
    const int B = in_sizes[0] / C_CLS;
    float* ws = (float*)d_ws;                    // B floats of per-row losses

    const int grid = (B < 1024) ? B : 1024;
    const int rpb  = B / grid;                   // rows per block (base)
    const int rem  = B % grid;                   // first `rem` blocks take +1
    clpl_rows_kernel<<<grid, BLOCK, 0, stream>>>(logits, mask, ws, rpb, rem);
    clpl_finalize_kernel<<<1, 256, 0, stream>>>(ws, (float*)d_out, B);
}